// DenseCRFFDR_2688649527544
// MI455X (gfx1250) — compile-verified
//
#include <hip/hip_runtime.h>

// ---------------------------------------------------------------------------
// Dense CRF-RNN (2 labels, 5 iters) on 64x128x128 volume for MI455X (gfx1250).
//
//  * ~44 MB working set << 192 MB L2: all intermediates live in d_ws and the
//    5 iterations stream from L2. Bilateral weights are recomputed on the fly
//    (TRANS-pipe exp) instead of materializing 108 MB of weight maps.
//  * Label einsums folded: pairwise = (compat@spatial_w)@sp
//    + (compat@bilateral_w)@bl; both constant 2x2 products computed by ONE
//    v_wmma_f32_16x16x4_f32 (exact f32) in a single-wave setup kernel.
//  * Hot 27-point bilateral gather is LDS-tiled: image/q0/q1 staged with
//    CDNA5 GLOBAL_LOAD_ASYNC_TO_LDS_B32 (ASYNCcnt) -> ~13x less global-side
//    traffic. Zero-filled halos exactly reproduce the reference's
//    zero-padded _shift3 semantics (OOB weight uses I_shift=0 and still
//    enters the denominator; OOB q contributes 0 to the numerator), so the
//    27-tap compute loop is branch-free.
// ---------------------------------------------------------------------------

#define D_ 64
#define H_ 128
#define W_ 128
#define N_ (D_ * H_ * W_) // 1048576 voxels

static constexpr float LOG_SQRT_2PI_F = 0.91893853320467274178f; // 0.5*ln(2*pi)

// 5-tap Gaussian (sigma=1), normalized (matches _k1d in reference)
static constexpr float K5_0 = 0.05448868454964294f; // |off| = 2
static constexpr float K5_1 = 0.24420134200323332f; // |off| = 1
static constexpr float K5_2 = 0.40261994689424740f; // center

// bilateral spatial factors exp(-d2 / (2*1.5^2)), d2 in {0,1,2,3}
static constexpr float SPW0 = 1.0f;
static constexpr float SPW1 = 0.80073740291680800f;
static constexpr float SPW2 = 0.64118038842995460f;
static constexpr float SPW3 = 0.51341711903259220f;

typedef __attribute__((ext_vector_type(2))) float v2f;
typedef __attribute__((ext_vector_type(8))) float v8f;

// Address-space-qualified pointer types matching the builtin's signature
// (diagnostic showed param 1 is a non-const addrspace(1) int*).
typedef __attribute__((address_space(1))) int as1_int;
typedef __attribute__((address_space(3))) int as3_int;

// ----- CDNA5 async-copy helpers (GLOBAL_LOAD_ASYNC_TO_LDS_B32 + ASYNCcnt) --
__device__ __forceinline__ void async_load_f32(const float* gptr, float* lptr) {
#if defined(__has_builtin) && __has_builtin(__builtin_amdgcn_global_load_async_to_lds_b32)
  int* g_generic = (int*)gptr;  // drop const / retype in generic AS
  int* l_generic = (int*)lptr;
  __builtin_amdgcn_global_load_async_to_lds_b32(
      (as1_int*)g_generic, (as3_int*)l_generic, /*offset=*/0, /*cpol=*/0);
#else
  const unsigned lds_off = (unsigned)(size_t)(as3_int*)(int*)lptr;
  asm volatile("global_load_async_to_lds_b32 %0, %1, off"
               :
               : "v"(lds_off), "v"(gptr)
               : "memory");
#endif
}

__device__ __forceinline__ void wait_async0() {
#if defined(__has_builtin) && __has_builtin(__builtin_amdgcn_s_wait_asynccnt)
  __builtin_amdgcn_s_wait_asynccnt(0);
#else
  asm volatile("s_wait_asynccnt 0x0" ::: "memory");
#endif
}

// ---------------------------------------------------------------------------
// Kernel 1: one wave, one v_wmma_f32_16x16x4_f32 computing
//   M1 = compat @ spatial_w   (D cols 0..1)
//   M2 = compat @ bilateral_w (D cols 2..3)
// ---------------------------------------------------------------------------
__global__ void crf_mats_kernel(const float* __restrict__ compat,
                                const float* __restrict__ spw,
                                const float* __restrict__ blw,
                                float* __restrict__ mats) {
  const int t = threadIdx.x; // 0..31, EXEC all ones at the WMMA
  v2f a;
  a.x = (t < 2) ? compat[t * 2 + 0] : 0.0f; // K=0 (lanes>=16 carry K=2 -> 0)
  a.y = (t < 2) ? compat[t * 2 + 1] : 0.0f; // K=1 (lanes>=16 carry K=3 -> 0)
  v2f b;
  float b0 = 0.0f, b1 = 0.0f;
  if (t < 2)      { b0 = spw[0 * 2 + t];       b1 = spw[1 * 2 + t]; }
  else if (t < 4) { b0 = blw[0 * 2 + (t - 2)]; b1 = blw[1 * 2 + (t - 2)]; }
  b.x = b0; // B row K=0
  b.y = b1; // B row K=1
  v8f c = {};
  v8f d = __builtin_amdgcn_wmma_f32_16x16x4_f32(
      false, a, false, b, (short)0, c, false, false);
  if (t < 4) {
    // mats layout: [M1 row-major x4, M2 row-major x4]
    const int i0 = (t < 2) ? t     : t + 2; // from D row 0
    const int i1 = (t < 2) ? t + 2 : t + 4; // from D row 1
    mats[i0] = d[0];
    mats[i1] = d[1];
  }
}

// ---------------------------------------------------------------------------
// Kernel 2: unary = h * (f1 - w0 - logprob), cur = h, and f1 passthrough.
// ---------------------------------------------------------------------------
__global__ __launch_bounds__(256) void crf_init_kernel(
    const float* __restrict__ image, const float* __restrict__ h,
    const float* __restrict__ f1, const float* __restrict__ w0,
    float* __restrict__ unary, float* __restrict__ cur,
    float* __restrict__ out_f1) {
  const int i = blockIdx.x * blockDim.x + threadIdx.x;
  if (i >= N_) return;
  const float img = image[i];
  const float logp = -0.5f * img * img - LOG_SQRT_2PI_F;
  const float f = f1[i];
  const float u = f - w0[0] - logp; // -(w0 + logprob - f1)
  const float h0 = h[i], h1 = h[N_ + i];
  unary[i]      = h0 * u;
  unary[N_ + i] = h1 * u;
  cur[i]      = h0;
  cur[N_ + i] = h1;
  out_f1[i] = f;
}

// ---------------------------------------------------------------------------
// Kernel 3: 2-channel softmax (stable), dst may be workspace q or d_out.
// ---------------------------------------------------------------------------
__global__ __launch_bounds__(256) void crf_softmax_kernel(
    const float* __restrict__ cur, float* __restrict__ dst) {
  const int i = blockIdx.x * blockDim.x + threadIdx.x;
  if (i >= N_) return;
  const float c0 = cur[i], c1 = cur[N_ + i];
  const float m = fmaxf(c0, c1);
  const float e0 = expf(c0 - m), e1 = expf(c1 - m);
  const float inv = 1.0f / (e0 + e1);
  dst[i]      = e0 * inv;
  dst[N_ + i] = e1 * inv;
}

// ---------------------------------------------------------------------------
// Kernel 4: one axis of the separable 5-tap blur, zero-padded.
// coordinate along axis = (i / stride) % len
// ---------------------------------------------------------------------------
__global__ __launch_bounds__(256) void crf_blur_axis_kernel(
    const float* __restrict__ src, float* __restrict__ dst,
    int stride, int len) {
  const int i = blockIdx.x * blockDim.x + threadIdx.x;
  if (i >= N_) return;
  const int c = (i / stride) % len;
#pragma unroll
  for (int ch = 0; ch < 2; ++ch) {
    const float* s = src + ch * N_;
    float acc = K5_2 * s[i];
    if (c >= 1)       acc += K5_1 * s[i - stride];
    if (c >= 2)       acc += K5_0 * s[i - 2 * stride];
    if (c + 1 < len)  acc += K5_1 * s[i + stride];
    if (c + 2 < len)  acc += K5_0 * s[i + 2 * stride];
    dst[ch * N_ + i] = acc;
  }
}

// ---------------------------------------------------------------------------
// Kernel 5: LDS-tiled 27-point bilateral message + combine:
//   cur = unary - (M1 @ sp + M2 @ bl)
// Tile 4x8x32 voxels (+1 halo) per 256-thread block; image/q0/q1 staged into
// LDS with async-to-LDS copies; zero halos reproduce zero-padded shifts.
// ---------------------------------------------------------------------------
__global__ __launch_bounds__(256) void crf_bilateral_kernel(
    const float* __restrict__ image, const float* __restrict__ q,
    const float* __restrict__ sp, const float* __restrict__ unary,
    const float* __restrict__ mats, float* __restrict__ cur) {
  constexpr int TZ = 4, TY = 8, TX = 32;
  constexpr int SZ = TZ + 2, SY = TY + 2, SX = TX + 2;
  constexpr int SN = SZ * SY * SX; // 2040 cells
  __shared__ float sI[SN];
  __shared__ float sQ0[SN];
  __shared__ float sQ1[SN];

  const int tid = threadIdx.x;
  const int z0 = blockIdx.z * TZ, y0 = blockIdx.y * TY, x0 = blockIdx.x * TX;

  // Stage (zero OOB cells with plain LDS stores; in-bounds cells via async
  // copies -- disjoint cell ownership, so no store/async race).
  for (int s = tid; s < SN; s += 256) {
    const int sz = s / (SY * SX);
    const int r  = s % (SY * SX);
    const int sy = r / SX;
    const int sx = r % SX;
    const int gz = z0 + sz - 1, gy = y0 + sy - 1, gx = x0 + sx - 1;
    const bool in = ((unsigned)gz < (unsigned)D_) &&
                    ((unsigned)gy < (unsigned)H_) &&
                    ((unsigned)gx < (unsigned)W_);
    if (in) {
      const int gi = (gz * H_ + gy) * W_ + gx;
      async_load_f32(image + gi,  &sI[s]);
      async_load_f32(q + gi,      &sQ0[s]);
      async_load_f32(q + N_ + gi, &sQ1[s]);
    } else {
      sI[s] = 0.0f;
      sQ0[s] = 0.0f;
      sQ1[s] = 0.0f;
    }
  }
  wait_async0();     // drain this wave's ASYNCcnt
  __syncthreads();   // all waves' stages visible

  const float m00 = mats[0], m01 = mats[1], m10 = mats[2], m11 = mats[3];
  const float b00 = mats[4], b01 = mats[5], b10 = mats[6], b11 = mats[7];
  constexpr float SPWT[4] = {SPW0, SPW1, SPW2, SPW3};

  const int lx = tid % TX;
  const int ly = tid / TX;
#pragma unroll
  for (int lz = 0; lz < TZ; ++lz) {
    const int ci = ((lz + 1) * SY + (ly + 1)) * SX + (lx + 1);
    const float Ic = sI[ci];
    float den = 0.0f, n0 = 0.0f, n1 = 0.0f;
#pragma unroll
    for (int dz = -1; dz <= 1; ++dz) {
#pragma unroll
      for (int dy = -1; dy <= 1; ++dy) {
#pragma unroll
        for (int dx = -1; dx <= 1; ++dx) {
          const int ni = ci + (dz * SY + dy) * SX + dx;
          const float In = sI[ni]; // 0 in halo-OOB cells (zero-padded shift)
          const float diff = Ic - In;
          const int d2 = dz * dz + dy * dy + dx * dx;
          const float w = SPWT[d2] * expf(-2.0f * diff * diff); // 2*0.5^2=0.5
          den += w;
          n0 += w * sQ0[ni]; // q halo = 0 -> matches zero-padded numerator
          n1 += w * sQ1[ni];
        }
      }
    }
    const float invden = 1.0f / den;
    const float bl0 = n0 * invden, bl1 = n1 * invden;
    const int i = ((z0 + lz) * H_ + (y0 + ly)) * W_ + (x0 + lx);
    const float sp0 = sp[i], sp1 = sp[N_ + i];
    cur[i]      = unary[i]      - (m00 * sp0 + m01 * sp1 + b00 * bl0 + b01 * bl1);
    cur[N_ + i] = unary[N_ + i] - (m10 * sp0 + m11 * sp1 + b10 * bl0 + b11 * bl1);
  }
}

// ---------------------------------------------------------------------------
// Launch. Workspace layout (floats):
//   [0,2N) cur  [2N,4N) q  [4N,6N) unary  [6N,8N) t1  [8N,10N) t2
//   [10N,10N+8) mats  -> ~42 MB total, all L2-resident.
// Output: d_out[0,2N) = final softmax(cur), d_out[2N,3N) = f1 passthrough.
// ---------------------------------------------------------------------------
extern "C" void kernel_launch(void* const* d_in, const int* in_sizes, int n_in,
                              void* d_out, int out_size, void* d_ws,
                              size_t ws_size, hipStream_t stream) {
  const float* image  = (const float*)d_in[0];
  const float* h      = (const float*)d_in[1];
  const float* f1     = (const float*)d_in[2];
  const float* w0     = (const float*)d_in[3];
  const float* spw    = (const float*)d_in[4];
  const float* blw    = (const float*)d_in[5];
  const float* compat = (const float*)d_in[6];

  float* out = (float*)d_out;
  float* ws  = (float*)d_ws;
  float* cur   = ws;
  float* q     = ws + 2 * (size_t)N_;
  float* unary = ws + 4 * (size_t)N_;
  float* t1    = ws + 6 * (size_t)N_;
  float* t2    = ws + 8 * (size_t)N_;
  float* mats  = ws + 10 * (size_t)N_;

  const dim3 blk(256);
  const dim3 grd((N_ + 255) / 256);
  const dim3 bil_grd(W_ / 32, H_ / 8, D_ / 4); // (4,16,16) tiles of 4x8x32

  crf_mats_kernel<<<1, 32, 0, stream>>>(compat, spw, blw, mats);
  crf_init_kernel<<<grd, blk, 0, stream>>>(image, h, f1, w0, unary, cur,
                                           out + 2 * (size_t)N_);
  for (int it = 0; it < 5; ++it) {
    crf_softmax_kernel<<<grd, blk, 0, stream>>>(cur, q);
    crf_blur_axis_kernel<<<grd, blk, 0, stream>>>(q, t1, H_ * W_, D_); // z
    crf_blur_axis_kernel<<<grd, blk, 0, stream>>>(t1, t2, W_, H_);     // y
    crf_blur_axis_kernel<<<grd, blk, 0, stream>>>(t2, t1, 1, W_);      // x
    crf_bilateral_kernel<<<bil_grd, blk, 0, stream>>>(image, q, t1, unary,
                                                      mats, cur);
  }
  crf_softmax_kernel<<<grd, blk, 0, stream>>>(cur, out);
}